// GNNInductive_62079457296460
// MI455X (gfx1250) — compile-verified
//
#include <hip/hip_runtime.h>
#include <hip/hip_bf16.h>

// GNN GraphConv x3 + linear readout for MI455X (gfx1250, wave32).
// Phase 1 (per layer): edge gather/scale/scatter-add  -- L2-resident (25.6MB
//   features + 19MB edge data << 192MB L2), f32 atomics at L2 atomic units.
// Phase 2 (per layer): D = relu(Agg*Wrel^T + X*Wroot^T + b) via
//   V_WMMA_F32_16X16X4_F32 (exact f32 semantics), weights staged in LDS.
// Phase 3: out = X*w_out^T + b_out, one wave per node, shfl_xor reduction.

#define HID 64
#define LDSW 66   // 64 + 2 pad floats -> kills LDS bank conflicts on column reads

typedef float v2f __attribute__((ext_vector_type(2)));
typedef float v8f __attribute__((ext_vector_type(8)));

// ---------------------------------------------------------------- zero agg
__global__ __launch_bounds__(256) void zero_kernel(float4* p, int n4) {
    int i = blockIdx.x * 256 + threadIdx.x;
    if (i < n4) p[i] = make_float4(0.f, 0.f, 0.f, 0.f);
}

// ------------------------------------------------- edge scatter (16 lanes/edge)
__global__ __launch_bounds__(256)
void edge_scatter_kernel(const float* __restrict__ x,
                         const int*   __restrict__ src_ids,
                         const int*   __restrict__ dst_ids,
                         const float* __restrict__ eattr,
                         float*       __restrict__ agg,
                         int n_edges) {
    int t = blockIdx.x * 256 + threadIdx.x;
    int e = t >> 4;                 // 16 lanes cooperate on one edge
    if (e >= n_edges) return;
    int c = (t & 15) * 4;           // 4 channels per lane (float4)

    // stream-prefetch future edge metadata (gfx1250 global_prefetch_b8)
    if (e + 4096 < n_edges) {
        __builtin_prefetch(&src_ids[e + 4096], 0, 1);
        __builtin_prefetch(&dst_ids[e + 4096], 0, 1);
    }

    int   src = src_ids[e];
    int   dst = dst_ids[e];
    float w   = eattr[e];

    float4 v = *(const float4*)(x + (long)src * HID + c);
    float* d = agg + (long)dst * HID + c;
    atomicAdd(d + 0, v.x * w);
    atomicAdd(d + 1, v.y * w);
    atomicAdd(d + 2, v.z * w);
    atomicAdd(d + 3, v.w * w);
}

// --------------------------------------------- fused dual-GEMM + bias + relu
// One wave per 16-node M-tile. N is assumed a multiple of 16 (100000 = 6250*16).
// A fragments (both matrices, all 16 K-steps) are preloaded into registers
// BEFORE any store, so xout may alias agg (in-place, halves workspace).
__global__ __launch_bounds__(256)
void gemm_relu_kernel(const float* agg,                     // [N,64]  (A1)
                      const float* __restrict__ xin,       // [N,64]  (A2)
                      const float* __restrict__ w_rel,     // [64,64] row-major [o][k]
                      const float* __restrict__ w_root,    // [64,64]
                      const float* __restrict__ b_rel,     // [64]
                      float*       xout,                   // [N,64]  (may alias agg)
                      int n_tiles) {
    __shared__ float lds_rel [HID * LDSW];
    __shared__ float lds_root[HID * LDSW];

    // cooperative coalesced staging of both weight matrices into LDS
    for (int i = threadIdx.x; i < HID * HID; i += 256) {
        int o = i >> 6, k = i & 63;
        lds_rel [o * LDSW + k] = w_rel[i];
        lds_root[o * LDSW + k] = w_root[i];
    }
    __syncthreads();

    int wave = blockIdx.x * 8 + (threadIdx.x >> 5);   // 8 wave32 per block
    if (wave >= n_tiles) return;                      // wave-uniform: EXEC stays all-1s
    int lane = threadIdx.x & 31;
    int row  = lane & 15;     // A row (M) == B/C column (N) == lane%16
    int half = lane >> 4;     // lane half selects K sub-pair / M upper half
    long base = (long)wave * 16;

    // Preload all A fragments: 16 K-steps of K=4; frag k-indices = 4*ks + 2*half + {0,1}
    v2f a_agg[16], a_x[16];
    const float* ar = agg + (base + row) * HID + 2 * half;
    const float* xr = xin + (base + row) * HID + 2 * half;
#pragma unroll
    for (int ks = 0; ks < 16; ++ks) {
        a_agg[ks] = *(const v2f*)(ar + 4 * ks);
        a_x  [ks] = *(const v2f*)(xr + 4 * ks);
    }

#pragma unroll
    for (int ot = 0; ot < 4; ++ot) {                  // 4 output-channel tiles of 16
        int ob = ot * 16;
        v8f c = {};
        const float* br = &lds_rel [(ob + row) * LDSW + 2 * half];
        const float* bo = &lds_root[(ob + row) * LDSW + 2 * half];
#pragma unroll
        for (int ks = 0; ks < 16; ++ks) {
            v2f fr = *(const v2f*)(br + 4 * ks);      // B[k][o] = W[o][k]
            v2f fo = *(const v2f*)(bo + 4 * ks);
            c = __builtin_amdgcn_wmma_f32_16x16x4_f32(false, a_agg[ks], false, fr,
                                                      (short)0, c, false, false);
            c = __builtin_amdgcn_wmma_f32_16x16x4_f32(false, a_x[ks],   false, fo,
                                                      (short)0, c, false, false);
        }
        float bias = b_rel[ob + row];                 // per-column bias
#pragma unroll
        for (int v = 0; v < 8; ++v) {                 // D: M = v + 8*half, N = row
            float r = c[v] + bias;
            r = r > 0.f ? r : 0.f;
            xout[(base + v + 8 * half) * HID + ob + row] = r;
        }
    }
}

// ------------------------------------------------------- readout 64 -> 1
__global__ __launch_bounds__(256)
void out_kernel(const float* __restrict__ x, const float* __restrict__ w_out,
                const float* __restrict__ b_out, float* __restrict__ out, int n) {
    int node = blockIdx.x * 8 + (threadIdx.x >> 5);   // one wave32 per node
    if (node >= n) return;
    int lane = threadIdx.x & 31;
    float2 xv = *(const float2*)(x + (long)node * HID + lane * 2);
    float2 wv = *(const float2*)(w_out + lane * 2);
    float p = xv.x * wv.x + xv.y * wv.y;
#pragma unroll
    for (int off = 16; off > 0; off >>= 1)
        p += __shfl_xor(p, off, 32);                  // wave32 reduction
    if (lane == 0) out[node] = p + b_out[0];
}

// ---------------------------------------------------------------- launcher
extern "C" void kernel_launch(void* const* d_in, const int* in_sizes, int n_in,
                              void* d_out, int out_size, void* d_ws, size_t ws_size,
                              hipStream_t stream) {
    const float* x      = (const float*)d_in[0];
    const int*   ei     = (const int*)  d_in[1];   // [2,E] int32 (harness convention)
    const float* ea     = (const float*)d_in[2];
    const float* w_rel  = (const float*)d_in[3];   // [3,64,64]
    const float* w_root = (const float*)d_in[4];   // [3,64,64]
    const float* b_rel  = (const float*)d_in[5];   // [3,64]
    const float* w_out  = (const float*)d_in[6];   // [1,64]
    const float* b_out  = (const float*)d_in[7];   // [1]

    int N = in_sizes[0] / HID;        // 100000
    int E = in_sizes[1] / 2;          // 1600000
    const int* srcs = ei;
    const int* dsts = ei + E;

    float* buf0 = (float*)d_ws;                 // ping-pong feature buffers
    float* buf1 = buf0 + (size_t)N * HID;       // 2 * 25.6 MB total

    int n4        = N * (HID / 4);
    int zero_blk  = (n4 + 255) / 256;
    int edge_blk  = (E * 16 + 255) / 256;
    int n_tiles   = (N + 15) / 16;              // N%16==0 here
    int gemm_blk  = (n_tiles + 7) / 8;
    int out_blk   = (N + 7) / 8;

    for (int l = 0; l < 3; ++l) {
        const float* xcur = (l == 0) ? x : (l == 1 ? buf0 : buf1);
        float*       buf  = (l == 1) ? buf1 : buf0;   // l0->buf0, l1->buf1, l2->buf0

        zero_kernel<<<zero_blk, 256, 0, stream>>>((float4*)buf, n4);
        edge_scatter_kernel<<<edge_blk, 256, 0, stream>>>(xcur, srcs, dsts, ea, buf, E);
        gemm_relu_kernel<<<gemm_blk, 256, 0, stream>>>(
            buf, xcur,
            w_rel  + (size_t)l * HID * HID,
            w_root + (size_t)l * HID * HID,
            b_rel  + (size_t)l * HID,
            buf, n_tiles);                            // in-place over agg
    }
    out_kernel<<<out_blk, 256, 0, stream>>>(buf0, w_out, b_out, (float*)d_out, N);
}